// CustomLoss_Z_70729521430605
// MI455X (gfx1250) — compile-verified
//
#include <hip/hip_runtime.h>

// ---------------------------------------------------------------------------
// CustomLoss for MI455X (gfx1250, wave32).
// Memory-bound streaming reductions; lanes map along contiguous k (64 floats)
// for fully coalesced 128B/wave loads. Whole footprint (134MB) fits in 192MB
// L2, so z is HBM-read once and L2-read twice. Wave-level reductions use
// V_WMMA_F32_16X16X4_F32 with an all-ones B matrix (one matrix op replaces a
// 5-step shuffle tree).
// ---------------------------------------------------------------------------

typedef __attribute__((ext_vector_type(2))) float v2f;
typedef __attribute__((ext_vector_type(8))) float v8f;

#define NX 256
#define NY 256
#define NK 64
#define CH_STRIDE (NX * NY * NK) // per-channel (targets) / per-batch (z) stride

__device__ __forceinline__ int zoff(int x, int y, int k) {
  return (x * NY + y) * NK + k; // offset within one (256,256,64) volume
}

// Sum of one value per lane across the full wave32 via WMMA.
// A (16x4 f32): lanes 0-15 VGPR0 -> A[L,0]=v, VGPR1 -> A[L,1]=0;
//               lanes 16-31 VGPR0 -> A[L-16,2]=v, VGPR1 -> A[L-16,3]=0.
// B = all ones (layout independent). D[m,n] = v_m + v_{m+16}.
// Sum of the 8 D VGPRs: lanes<16 get rows 0-7, lanes>=16 get rows 8-15;
// one xor-16 shuffle combines the halves. Requires EXEC == all ones.
__device__ __forceinline__ float wave_sum32(float v) {
  v2f a;
  a[0] = v;
  a[1] = 0.0f;
  v2f b;
  b[0] = 1.0f;
  b[1] = 1.0f;
  v8f c = {0.f, 0.f, 0.f, 0.f, 0.f, 0.f, 0.f, 0.f};
  v8f d = __builtin_amdgcn_wmma_f32_16x16x4_f32(false, a, false, b, (short)0,
                                                c, false, false);
  float s = d[0] + d[1] + d[2] + d[3] + d[4] + d[5] + d[6] + d[7];
  s += __shfl_xor(s, 16, 32);
  return s;
}

// Block (64,4) = 8 waves -> single float partial per block.
__device__ __forceinline__ void block_reduce_store(float acc, float* dst) {
  __shared__ float wpart[8];
  int ltid = threadIdx.y * 64 + threadIdx.x;
  int wid = ltid >> 5;
  int lane = ltid & 31;
  float t = wave_sum32(acc);
  if (lane == 0) wpart[wid] = t;
  __syncthreads();
  if (wid == 0) { // whole wave 0 -> EXEC all ones for the WMMA
    float v = (lane < 8) ? wpart[lane] : 0.0f;
    float tot = wave_sum32(v);
    if (lane == 0) *dst = tot;
  }
}

// ---------------------------------------------------------------------------
// K1: per-(b,k) sum & sumsq of dz over (x,y), partial per y-group of 4 rows.
// grid (64 ygroups, 2 b), block (64 k-lanes, 4 y).
// ---------------------------------------------------------------------------
__global__ __launch_bounds__(256) void k_std(const float* __restrict__ z,
                                             float* __restrict__ sumP,
                                             float* __restrict__ sqP) {
  int k = threadIdx.x;
  int y = blockIdx.x * 4 + threadIdx.y;
  int b = blockIdx.y;
  float s = 0.f, q = 0.f;
  if (k < NK - 1) {
    const float* zz = z + (size_t)b * CH_STRIDE;
    int base = zoff(0, y, k);
    for (int x = 0; x < NX; ++x) {
      float d = zz[base + 1] - zz[base];
      s += d;
      q += d * d;
      base += NY * NK;
    }
  }
  __shared__ float ss[4][64];
  __shared__ float sq[4][64];
  ss[threadIdx.y][k] = s;
  sq[threadIdx.y][k] = q;
  __syncthreads();
  if (threadIdx.y == 0 && k < NK - 1) {
    float S = ss[0][k] + ss[1][k] + ss[2][k] + ss[3][k];
    float Q = sq[0][k] + sq[1][k] + sq[2][k] + sq[3][k];
    int p = (b * 64 + (int)blockIdx.x) * (NK - 1) + k;
    sumP[p] = S;
    sqP[p] = Q;
  }
}

// ---------------------------------------------------------------------------
// K2: sum of lap^2. lap element domain (2,254,254,61).
// grid (8 x-chunks, 64 ygroups, 2 b), block (64,4).
// ---------------------------------------------------------------------------
__global__ __launch_bounds__(256) void k_smooth(const float* __restrict__ z,
                                                float* __restrict__ part) {
  int kk = threadIdx.x;                  // valid < 61
  int yy = blockIdx.y * 4 + threadIdx.y; // valid < 254
  int b = blockIdx.z;
  float acc = 0.f;
  if (kk < NK - 3 && yy < NY - 2) {
    const float* zz = z + (size_t)b * CH_STRIDE;
    int x0 = blockIdx.x * 32;
    int xe = min(x0 + 32, NX - 2);
    for (int xx = x0; xx < xe; ++xx) {
      const float* pc = zz + zoff(xx + 1, yy + 1, kk);
      float c0 = pc[0], c1 = pc[1], c2 = pc[2], c3 = pc[3];
      float dck = c1 - c0;  // dz2 at kk
      float dc1 = c2 - c1;  // dz2 at kk+1 (lap center)
      float dc2 = c3 - c2;  // dz2 at kk+2
      const float* pxm = zz + zoff(xx, yy + 1, kk + 1);
      const float* pxp = zz + zoff(xx + 2, yy + 1, kk + 1);
      const float* pym = zz + zoff(xx + 1, yy, kk + 1);
      const float* pyp = zz + zoff(xx + 1, yy + 2, kk + 1);
      float dxm = pxm[1] - pxm[0];
      float dxp = pxp[1] - pxp[0];
      float dym = pym[1] - pym[0];
      float dyp = pyp[1] - pyp[0];
      float lap = 6.f * (dc1 * dc1) - dxm * dxm - dxp * dxp - dym * dym -
                  dyp * dyp - dc2 * dc2 - dck * dck;
      acc += lap * lap;
    }
  }
  int bid = ((int)blockIdx.z * gridDim.y + blockIdx.y) * gridDim.x + blockIdx.x;
  block_reduce_store(acc, part + bid);
}

// ---------------------------------------------------------------------------
// K3: sum of num^2/den over (2,255,255,63). Corner cubes of z,bx,by,bz with
// x-rolling registers (x+1 corners become next iteration's x corners).
// grid (8 x-chunks, 64 ygroups, 2 b), block (64,4).
// ---------------------------------------------------------------------------
__global__ __launch_bounds__(256) void k_div(const float* __restrict__ z,
                                             const float* __restrict__ tg,
                                             float* __restrict__ part) {
  int k = threadIdx.x;                  // valid < 63
  int y = blockIdx.y * 4 + threadIdx.y; // valid < 255
  int b = blockIdx.z;
  float acc = 0.f;
  if (k < NK - 1 && y < NY - 1) {
    const float* zz = z + (size_t)b * CH_STRIDE;
    const float* bx = tg + (size_t)(b * 3 + 0) * CH_STRIDE;
    const float* by = tg + (size_t)(b * 3 + 1) * CH_STRIDE;
    const float* bz = tg + (size_t)(b * 3 + 2) * CH_STRIDE;
    int x0 = blockIdx.x * 32;
    int xe = min(x0 + 32, NX - 1);
    int o00 = zoff(x0, y, k);
    int o10 = o00 + NK; // (x0, y+1, k)
    // Corner regs: A=(i=0,j=0) B=(i=0,j=1) C=(i=1,j=0) D=(i=1,j=1); suffix=kc.
    float zA0 = zz[o00], zA1 = zz[o00 + 1], zB0 = zz[o10], zB1 = zz[o10 + 1];
    float xA0 = bx[o00], xA1 = bx[o00 + 1], xB0 = bx[o10], xB1 = bx[o10 + 1];
    float yA0 = by[o00], yA1 = by[o00 + 1], yB0 = by[o10], yB1 = by[o10 + 1];
    float wA0 = bz[o00], wA1 = bz[o00 + 1], wB0 = bz[o10], wB1 = bz[o10 + 1];
    for (int x = x0; x < xe; ++x) {
      int n00 = zoff(x + 1, y, k);
      int n10 = n00 + NK;
      float zC0 = zz[n00], zC1 = zz[n00 + 1], zD0 = zz[n10], zD1 = zz[n10 + 1];
      float xC0 = bx[n00], xC1 = bx[n00 + 1], xD0 = bx[n10], xD1 = bx[n10 + 1];
      float yC0 = by[n00], yC1 = by[n00 + 1], yD0 = by[n10], yD1 = by[n10 + 1];
      float wC0 = bz[n00], wC1 = bz[n00 + 1], wD0 = bz[n10], wD1 = bz[n10 + 1];

      float azA = fabsf(zA1 - zA0); // (0,0,1)-(0,0,0)
      float azB = fabsf(zB1 - zB0); // (0,1,1)-(0,1,0)
      float azC = fabsf(zC1 - zC0); // (1,0,1)-(1,0,0)
      float azD = fabsf(zD1 - zD0); // (1,1,1)-(1,1,0)
      const float SIXTH = 1.f / 6.f;
      float num = 0.25f * (xC0 + xD0 + xC1 + xD1) * 0.5f * (azC + azD) -
                  0.25f * (xA0 + xB0 + xA1 + xB1) * 0.5f * (azA + azB) +
                  0.25f * (yB0 + yD0 + yB1 + yD1) * 0.5f * (azB + azD) -
                  0.25f * (yA0 + yC0 + yA1 + yC1) * 0.5f * (azA + azC) +
                  0.25f * (wA1 + wB1 + wC1 + wD1) -
                  0.25f * (wA0 + wB0 + wC0 + wD0) +
                  (xA1 + xC1 + xD1) * (zA1 - zC1) * SIXTH +
                  (xB1 + xD1 + xC1) * (zB1 - zD1) * SIXTH +
                  (yC1 + yD1 + yB1) * (zC1 - zD1) * SIXTH +
                  (yA1 + yB1 + yD1) * (zA1 - zB1) * SIXTH -
                  (xA0 + xC0 + xD0) * (zA0 - zC0) * SIXTH -
                  (xB0 + xD0 + xC0) * (zB0 - zD0) * SIXTH -
                  (yC0 + yD0 + yB0) * (zC0 - zD0) * SIXTH -
                  (yA0 + yB0 + yD0) * (zA0 - zB0) * SIXTH;
      float sbx = xA0 + xA1 + xB0 + xB1 + xC0 + xC1 + xD0 + xD1;
      float sby = yA0 + yA1 + yB0 + yB1 + yC0 + yC1 + yD0 + yD1;
      float sbz = wA0 + wA1 + wB0 + wB1 + wC0 + wC1 + wD0 + wD1;
      float den = (0.125f * sbx) * (0.125f * sbx) +
                  (0.125f * sby) * (0.125f * sby) +
                  (0.125f * sbz) * (0.125f * sbz) + 1e-10f;
      acc += num * num / den;

      zA0 = zC0; zA1 = zC1; zB0 = zD0; zB1 = zD1;
      xA0 = xC0; xA1 = xC1; xB0 = xD0; xB1 = xD1;
      yA0 = yC0; yA1 = yC1; yB0 = yD0; yB1 = yD1;
      wA0 = wC0; wA1 = wC1; wB0 = wD0; wB1 = wD1;
    }
  }
  int bid = ((int)blockIdx.z * gridDim.y + blockIdx.y) * gridDim.x + blockIdx.x;
  block_reduce_store(acc, part + bid);
}

// ---------------------------------------------------------------------------
// K4: one wave, double-precision combine of all partials -> 2 output floats.
// ---------------------------------------------------------------------------
__global__ __launch_bounds__(32) void k_finalize(
    const float* __restrict__ smoothP, int nSmooth,
    const float* __restrict__ divP, int nDiv, const float* __restrict__ sumP,
    const float* __restrict__ sqP, float* __restrict__ out) {
  int lane = threadIdx.x;
  double aS = 0.0, aD = 0.0, aStd = 0.0;
  for (int i = lane; i < nSmooth; i += 32) aS += (double)smoothP[i];
  for (int i = lane; i < nDiv; i += 32) aD += (double)divP[i];
  for (int t = lane; t < 2 * (NK - 1); t += 32) {
    int b = t / (NK - 1);
    int k = t % (NK - 1);
    double s = 0.0, q = 0.0;
    for (int g = 0; g < 64; ++g) {
      int p = (b * 64 + g) * (NK - 1) + k;
      s += (double)sumP[p];
      q += (double)sqP[p];
    }
    const double N = 65536.0;
    double var = (q - s * s / N) / (N - 1.0);
    aStd += sqrt(var > 0.0 ? var : 0.0);
  }
  for (int m = 16; m > 0; m >>= 1) {
    aS += __shfl_xor(aS, m, 32);
    aD += __shfl_xor(aD, m, 32);
    aStd += __shfl_xor(aStd, m, 32);
  }
  if (lane == 0) {
    double loss_smooth = aS / 7870952.0;  // 2*254*254*61
    double loss_div = aD / 8193150.0;     // 2*255*255*63
    double loss_std = aStd / 126.0;
    out[0] = (float)(loss_div * 1.0e9);                       // W_DIV
    out[1] = (float)(loss_smooth * 10.0 + loss_std * 100.0);  // W_SMOOTH/W_STD
  }
}

extern "C" void kernel_launch(void* const* d_in, const int* in_sizes, int n_in,
                              void* d_out, int out_size, void* d_ws,
                              size_t ws_size, hipStream_t stream) {
  (void)in_sizes;
  (void)n_in;
  (void)out_size;
  (void)ws_size;
  const float* z = (const float*)d_in[0];  // outputs (2,1,256,256,64)
  const float* tg = (const float*)d_in[1]; // targets (2,3,256,256,64)
  float* out = (float*)d_out;

  // Workspace layout (floats), ~71KB; every slot rewritten each launch.
  float* w = (float*)d_ws;
  float* smoothP = w;         // 1024 block partials
  float* divP = w + 1024;     // 1024 block partials
  float* sumP = w + 2048;     // 2*64*63 = 8064 per-(b,ygroup,k) dz sums
  float* sqP = w + 2048 + 8064; // 8064 dz^2 sums

  dim3 blk(64, 4, 1); // 8 wave32s; lanes along contiguous k
  k_std<<<dim3(64, 2, 1), blk, 0, stream>>>(z, sumP, sqP);
  k_smooth<<<dim3(8, 64, 2), blk, 0, stream>>>(z, smoothP);
  k_div<<<dim3(8, 64, 2), blk, 0, stream>>>(z, tg, divP);
  k_finalize<<<1, 32, 0, stream>>>(smoothP, 1024, divP, 1024, sumP, sqP, out);
}